// CrossEntropyLossWithGaussianSmoothedLabels_74096775790992
// MI455X (gfx1250) — compile-verified
//
#include <hip/hip_runtime.h>
#include <math.h>

// CrossEntropy with Gaussian-smoothed labels, closed-form labels:
//   label[c] = 1.0 (c==t); exp(-2^d/4) for d=|c-t| in 1..3 (in-bounds); else 0
//   loss_row = S(t)*lse(row) - sum_c label[c]*pred[c],   out = mean over rows
//
// B=16, T=2048, C=722.  Memory-bound: 94.6MB @ 23.3TB/s ~= 4us.
// One wave32 per row; row held in registers (12 float2 per lane).
// Cross-lane sums via V_WMMA_F32_16X16X4_F32 (exact fp32, ones-B).

#define NUM_CLASSES 722
#define NUM_PAIRS   (NUM_CLASSES / 2)   // 361 float2 per row
#define NUM_ROWS    (16 * 2048)         // 32768
#define BLOCKS      512
#define THREADS     256                 // 8 waves per block
#define WAVES_PER_BLOCK (THREADS / 32)

#define W1 0.60653065971263342360f   // exp(-0.5)
#define W2 0.36787944117144232160f   // exp(-1.0)
#define W3 0.13533528323661269189f   // exp(-2.0)

typedef float v2f __attribute__((ext_vector_type(2)));
typedef float v8f __attribute__((ext_vector_type(8)));

__device__ __forceinline__ float wave_max32(float v) {
    #pragma unroll
    for (int o = 16; o > 0; o >>= 1)
        v = fmaxf(v, __shfl_xor(v, o, 32));
    return v;
}

// Exact fp32 sum of one value per lane across the wave, via WMMA.
// A (16x4 f32, 2 VGPRs): VGPR0 = x (lanes0-15 -> A[m][0], lanes16-31 -> A[m][2]),
// VGPR1 = 0. B (4x16 f32, 2 VGPRs) = all ones.
// D[m][n] = x[m] + x[m+16] for every n. Sum over m: per-lane add of the 8 D
// VGPRs gives rows 0-7 (lanes 0-15) / rows 8-15 (lanes 16-31) column sums;
// adding the opposite half via shfl_xor(16) yields the full wave sum in all lanes.
__device__ __forceinline__ float wave_sum_wmma(float x) {
    v2f a; a[0] = x;    a[1] = 0.0f;
    v2f b; b[0] = 1.0f; b[1] = 1.0f;
    v8f c = {};
    c = __builtin_amdgcn_wmma_f32_16x16x4_f32(
            /*neg_a=*/false, a, /*neg_b=*/false, b,
            /*c_mod=*/(short)0, c, /*reuse_a=*/false, /*reuse_b=*/false);
    float p = ((c[0] + c[1]) + (c[2] + c[3])) + ((c[4] + c[5]) + (c[6] + c[7]));
    return p + __shfl_xor(p, 16, 32);
}

__device__ __forceinline__ float label_w(int d) {
    int a = d < 0 ? -d : d;
    if (d == 0) return 1.0f;
    if (a == 1) return W1;
    if (a == 2) return W2;
    if (a == 3) return W3;
    return 0.0f;
}

__global__ void __launch_bounds__(THREADS)
ce_gauss_rows_kernel(const float* __restrict__ pred,
                     const int*   __restrict__ target,
                     float* __restrict__ block_partials) {
    const int lane  = threadIdx.x & 31;
    const int wave  = threadIdx.x >> 5;
    const int gwave = blockIdx.x * WAVES_PER_BLOCK + wave;   // 0..4095
    const int nwave = gridDim.x * WAVES_PER_BLOCK;           // 4096 -> 8 rows/wave

    __shared__ float s_wave[WAVES_PER_BLOCK];

    float acc = 0.0f;

    // Uniform trip count across all waves -> EXEC is all-ones at every WMMA.
    for (int row = gwave; row < NUM_ROWS; row += nwave) {
        const float2* rp = (const float2*)(pred + (size_t)row * NUM_CLASSES);
        const int t = target[row];

        // ---- pass 1: load row into registers, lane-local max ----
        float2 v[12];
        float m = -INFINITY;
        #pragma unroll
        for (int i = 0; i < 12; ++i) {
            const int p2 = i * 32 + lane;            // float2 index in row
            if (p2 < NUM_PAIRS) {
                v[i] = rp[p2];                       // coalesced global_load_b64
            } else {
                v[i].x = -INFINITY; v[i].y = -INFINITY;
            }
            m = fmaxf(m, fmaxf(v[i].x, v[i].y));
        }
        m = wave_max32(m);                           // uniform across wave

        // ---- pass 2 (registers): sum exp(v-m) and label-weighted dot ----
        float se = 0.0f, dt = 0.0f;
        #pragma unroll
        for (int i = 0; i < 12; ++i) {
            const int c0 = (i * 32 + lane) * 2;
            se += __expf(v[i].x - m) + __expf(v[i].y - m);  // expf(-inf)=0 pads
            if (c0 < NUM_CLASSES)     dt += label_w(c0     - t) * v[i].x;
            if (c0 + 1 < NUM_CLASSES) dt += label_w(c0 + 1 - t) * v[i].y;
        }

        // ---- exact fp32 wave reductions via WMMA ----
        const float se_t = wave_sum_wmma(se);
        const float dt_t = wave_sum_wmma(dt);

        // analytic label mass S(t)
        float S = 1.0f;
        S += W1 * (float)((t - 1 >= 0) + (t + 1 < NUM_CLASSES));
        S += W2 * (float)((t - 2 >= 0) + (t + 2 < NUM_CLASSES));
        S += W3 * (float)((t - 3 >= 0) + (t + 3 < NUM_CLASSES));

        const float lse = m + __logf(se_t);
        acc += S * lse - dt_t;                       // uniform across wave
    }

    if (lane == 0) s_wave[wave] = acc;
    __syncthreads();
    if (threadIdx.x == 0) {
        float b = 0.0f;
        #pragma unroll
        for (int w = 0; w < WAVES_PER_BLOCK; ++w) b += s_wave[w];
        block_partials[blockIdx.x] = b;
    }
}

__global__ void __launch_bounds__(THREADS)
ce_gauss_final_reduce(const float* __restrict__ block_partials,
                      int n, float* __restrict__ out) {
    const int lane = threadIdx.x & 31;
    const int wave = threadIdx.x >> 5;
    __shared__ float s_wave[WAVES_PER_BLOCK];

    float s = 0.0f;
    for (int i = threadIdx.x; i < n; i += THREADS) s += block_partials[i];
    #pragma unroll
    for (int o = 16; o > 0; o >>= 1) s += __shfl_xor(s, o, 32);

    if (lane == 0) s_wave[wave] = s;
    __syncthreads();
    if (threadIdx.x == 0) {
        float tot = 0.0f;
        #pragma unroll
        for (int w = 0; w < WAVES_PER_BLOCK; ++w) tot += s_wave[w];
        out[0] = tot / (float)NUM_ROWS;
    }
}

extern "C" void kernel_launch(void* const* d_in, const int* in_sizes, int n_in,
                              void* d_out, int out_size, void* d_ws, size_t ws_size,
                              hipStream_t stream) {
    (void)in_sizes; (void)n_in; (void)out_size; (void)ws_size;
    const float* pred   = (const float*)d_in[0];   // (16,2048,722) fp32
    const int*   target = (const int*)  d_in[1];   // (16,2048) int
    float* out      = (float*)d_out;               // scalar fp32
    float* partials = (float*)d_ws;                // BLOCKS floats of scratch

    ce_gauss_rows_kernel<<<BLOCKS, THREADS, 0, stream>>>(pred, target, partials);
    ce_gauss_final_reduce<<<1, THREADS, 0, stream>>>(partials, BLOCKS, out);
}